// TopK_19576460935400
// MI455X (gfx1250) — compile-verified
//
#include <hip/hip_runtime.h>
#include <stdint.h>

// Top-k row masking: out = x * (mask of top-64 per 4096-wide row).
// Memory-bound: 512MB traffic -> ~22us floor @ 23.3 TB/s.
// One 256-thread block per row; async global->LDS staging (CDNA5 ASYNC path),
// radix-256 select (4 passes, LDS histogram) for the exact 64th value.
// Bin scans done by a single wave with __shfl (no barriers inside),
// ~14 block barriers total instead of ~100.

#define F    4096
#define TPB  256
#define EPT  16   // elements per thread (contiguous chunk -> index order == (tid, j) order)

__global__ __launch_bounds__(TPB) void topk_mask_kernel(
    const float* __restrict__ x, const int* __restrict__ kptr,
    float* __restrict__ out, int nrows)
{
    const int row = blockIdx.x;
    const int t   = threadIdx.x;
    const float* xr   = x   + (size_t)row * F;
    float*       outr = out + (size_t)row * F;

    __shared__ __align__(16) float tile[F];   // 16 KB row stage
    __shared__ unsigned hist[TPB];            // 256-bin histogram / scan buffer
    __shared__ unsigned s_bin, s_above;

    // ---- async global -> LDS copy of this thread's 64B chunk (gfx1250 ASYNC path) ----
    // Low 32 bits of a generic shared pointer are the LDS byte address.
    unsigned ldsbase = (unsigned)(unsigned long long)(uintptr_t)&tile[0];
#pragma unroll
    for (int i = 0; i < 4; ++i) {
        unsigned laddr = ldsbase + (unsigned)(t * 64 + i * 16);
        unsigned long long gaddr =
            (unsigned long long)(uintptr_t)(xr + t * EPT + i * 4);
        asm volatile("global_load_async_to_lds_b128 %0, %1, off"
                     :: "v"(laddr), "v"(gaddr) : "memory");
    }
    asm volatile("s_wait_asynccnt 0x0" ::: "memory");

    // ---- pull chunk into registers (ds_load_b128), build monotonic u32 keys ----
    float    vals[EPT];
    unsigned keys[EPT];
#pragma unroll
    for (int i = 0; i < 4; ++i) {
        const float4 v = *reinterpret_cast<const float4*>(&tile[t * EPT + i * 4]);
        vals[i*4+0] = v.x; vals[i*4+1] = v.y; vals[i*4+2] = v.z; vals[i*4+3] = v.w;
    }
#pragma unroll
    for (int j = 0; j < EPT; ++j) {
        unsigned b = __float_as_uint(vals[j]);
        // descending float order == descending key order
        keys[j] = b ^ (((unsigned)((int)b >> 31)) | 0x80000000u);
    }

    const int K = kptr[0];     // k == 64 (read on device; works for any k <= F)
    int rem = K;               // how many still needed from current bin downward
    unsigned prefix = 0;       // selected high bytes of the threshold key
    bool act[EPT];
#pragma unroll
    for (int j = 0; j < EPT; ++j) act[j] = true;

    // ---- 4-pass radix-256 select (MSB first) ----
    for (int pass = 0; pass < 4; ++pass) {
        const int shift = 24 - 8 * pass;

        hist[t] = 0u;
        __syncthreads();
#pragma unroll
        for (int j = 0; j < EPT; ++j)
            if (act[j]) atomicAdd(&hist[(keys[j] >> shift) & 0xFFu], 1u);
        __syncthreads();

        // Wave 0 alone: suffix sums over 256 bins (8 bins/lane) + crossing-bin
        // selection, entirely with registers + __shfl (no barriers inside).
        if (t < 32) {
            const int l = t;
            const unsigned remu = (unsigned)rem;
            unsigned c[8];
#pragma unroll
            for (int i = 0; i < 8; ++i) c[i] = hist[l * 8 + i];
            unsigned suf[8];
            suf[7] = c[7];
#pragma unroll
            for (int i = 6; i >= 0; --i) suf[i] = suf[i + 1] + c[i];
            const unsigned laneTotal = suf[0];
            unsigned acc = laneTotal;               // -> sum over lanes >= l
#pragma unroll
            for (int off = 1; off < 32; off <<= 1) {
                unsigned v = __shfl_down(acc, off, 32);
                if (l + off < 32) acc += v;
            }
            const unsigned above = acc - laneTotal; // sum over lanes > l
#pragma unroll
            for (int i = 0; i < 8; ++i) {
                unsigned sufG     = suf[i] + above;                      // bins >= 8l+i
                unsigned sufGnext = (i < 7) ? (suf[i + 1] + above) : above; // bins > 8l+i
                if (sufG >= remu && sufGnext < remu) {                   // unique crossing
                    s_bin   = (unsigned)(l * 8 + i);
                    s_above = sufGnext;
                }
            }
        }
        __syncthreads();

        const unsigned bin = s_bin;
        rem   -= (int)s_above;            // strictly-greater elements are locked in
        prefix = (prefix << 8) | bin;
        if (pass < 3) {
#pragma unroll
            for (int j = 0; j < EPT; ++j)
                act[j] = act[j] && (((keys[j] >> shift) & 0xFFu) == bin);
        }
    }

    const unsigned T = prefix;            // exact k-th largest key
    // rem == number of (key == T) elements to keep, >= 1

    // ---- tie-break: rank equal-to-T elements in ascending index order ----
    unsigned eq = 0;
#pragma unroll
    for (int j = 0; j < EPT; ++j) eq += (keys[j] == T) ? 1u : 0u;
    hist[t] = eq;
    __syncthreads();
    // Wave 0: exclusive prefix over the 256 per-thread counts (8/lane + __shfl_up).
    if (t < 32) {
        const int l = t;
        unsigned c[8];
#pragma unroll
        for (int i = 0; i < 8; ++i) c[i] = hist[l * 8 + i];
        unsigned p[8];
        p[0] = c[0];
#pragma unroll
        for (int i = 1; i < 8; ++i) p[i] = p[i - 1] + c[i];
        const unsigned laneTotal = p[7];
        unsigned acc = laneTotal;               // -> sum over lanes <= l
#pragma unroll
        for (int off = 1; off < 32; off <<= 1) {
            unsigned v = __shfl_up(acc, off, 32);
            if (l >= off) acc += v;
        }
        const unsigned excl = acc - laneTotal;  // sum over lanes < l
#pragma unroll
        for (int i = 0; i < 8; ++i)
            hist[l * 8 + i] = excl + (i ? p[i - 1] : 0u);
    }
    __syncthreads();
    const unsigned base = hist[t];              // exclusive prefix for this thread

    // ---- masked write (global_store_b128) ----
    unsigned r = 0;
    float ovals[EPT];
#pragma unroll
    for (int j = 0; j < EPT; ++j) {
        const unsigned key = keys[j];
        bool keep;
        if (key > T)       keep = true;
        else if (key == T) { keep = (base + r) < (unsigned)rem; ++r; }
        else               keep = false;
        ovals[j] = keep ? vals[j] : 0.0f;
    }
#pragma unroll
    for (int i = 0; i < 4; ++i) {
        float4 v;
        v.x = ovals[i*4+0]; v.y = ovals[i*4+1];
        v.z = ovals[i*4+2]; v.w = ovals[i*4+3];
        *reinterpret_cast<float4*>(&outr[t * EPT + i * 4]) = v;
    }
}

extern "C" void kernel_launch(void* const* d_in, const int* in_sizes, int n_in,
                              void* d_out, int out_size, void* d_ws, size_t ws_size,
                              hipStream_t stream) {
    const float* x = (const float*)d_in[0];
    const int*   k = (const int*)d_in[1];
    float*     out = (float*)d_out;
    const int nrows = in_sizes[0] / F;
    hipLaunchKernelGGL(topk_mask_kernel, dim3(nrows), dim3(TPB), 0, stream,
                       x, k, out, nrows);
}